// SSIM_25812753449800
// MI455X (gfx1250) — compile-verified
//
#include <hip/hip_runtime.h>

// ---------------------------------------------------------------------------
// SSIM on MI455X (gfx1250), fused single pass.
//  - bandwidth-bound (201 MB in, 4 B out -> ~8.6 us roofline @ 23.3 TB/s)
//  - separable 11-tap Gaussian conv done as banded-matrix GEMMs on the
//    full-precision f32 WMMA path (V_WMMA_F32_16X16X4_F32); f16/bf16 WMMA
//    would destroy the sigma = E[x^2]-E[x]^2 cancellation (x^2 up to ~55k).
//  - per block (256 thr = 8 wave32): 64x32 output region, luma footprint in
//    LDS; each wave does one 16x16 tile = 105 WMMA ops (5 maps x 21).
//  - all bounds handling is branch-free (clamped index + select-to-zero) so
//    the wave stays convergent through the WMMA stream: no exec toggling,
//    unconditional ds_load_2addr feeds.
// ---------------------------------------------------------------------------

typedef float v2f __attribute__((ext_vector_type(2)));
typedef float v8f __attribute__((ext_vector_type(8)));

// Normalized 11-tap Gaussian, sigma = 1.5 (matches reference _make_window)
__constant__ float GW[11] = {
    0.00102840f, 0.00759876f, 0.03600075f, 0.10936077f, 0.21300561f,
    0.26601169f, 0.21300561f, 0.10936077f, 0.03600075f, 0.00759876f,
    0.00102840f};

#define SSIM_C1 6.5025f
#define SSIM_C2 58.5225f

__device__ __forceinline__ v8f wmma4_f32(v2f a, v2f b, v8f c) {
  // D = A(16x4,f32) * B(4x16,f32) + C(16x16,f32)
  return __builtin_amdgcn_wmma_f32_16x16x4_f32(
      /*neg_a=*/false, a, /*neg_b=*/false, b,
      /*c_mod=*/(short)0, c, /*reuse_a=*/false, /*reuse_b=*/false);
}

__device__ __forceinline__ float luma_of(float r, float g, float b) {
  float rr = (r + 1.0f) * 127.5f;
  float gg = (g + 1.0f) * 127.5f;
  float bb = (b + 1.0f) * 127.5f;
  return (rr * 65.738f + gg * 129.057f + bb * 25.064f) * (1.0f / 256.0f) +
         16.0f;
}

__device__ __forceinline__ float map_val(int m, float y1, float y2) {
  switch (m) {
    case 0: return y1;
    case 1: return y2;
    case 2: return y1 * y1;
    case 3: return y2 * y2;
    default: return y1 * y2;
  }
}

__global__ void ssim_zero_acc(double* acc) {
  if (threadIdx.x == 0 && blockIdx.x == 0) acc[0] = 0.0;
}

__global__ __launch_bounds__(256) void ssim_main_kernel(
    const float* __restrict__ img1, const float* __restrict__ img2,
    double* __restrict__ acc) {
  // Region: 64 rows x 32 cols of output; footprint 74 x 42 luma pixels.
  __shared__ float Y1s[74][44];  // stride 44 to dodge bank conflicts
  __shared__ float Y2s[74][44];
  __shared__ float Vb[8][16][32];  // per-wave stage-V scratch (16x28 used)

  const int b  = blockIdx.z;
  const int Ry = blockIdx.y * 64;
  const int Rx = blockIdx.x * 32;
  const int tid = threadIdx.x;

  // ---- Phase 1: cooperative luma load into LDS (coalesced RGB reads).
  // Branch-free: clamp coordinates into the image; out-of-image footprint
  // pixels only ever feed outputs that are masked from the final sum.
  for (int idx = tid; idx < 74 * 42; idx += 256) {
    int fr = idx / 42, fc = idx % 42;
    int gr = min(Ry + fr, 511), gc = min(Rx + fc, 511);
    size_t base = (((size_t)b * 512 + (size_t)gr) * 512 + (size_t)gc) * 3;
    Y1s[fr][fc] = luma_of(img1[base + 0], img1[base + 1], img1[base + 2]);
    Y2s[fr][fc] = luma_of(img2[base + 0], img2[base + 1], img2[base + 2]);
  }
  __syncthreads();

  // ---- Phase 2: one 16x16 output tile per wave, all conv as f32 WMMA ----
  const int lane = tid & 31;
  const int w    = tid >> 5;        // wave 0..7
  const int r0   = (w >> 1) * 16;   // tile row origin in region
  const int c0   = (w & 1) * 16;    // tile col origin in region
  const int hi   = lane >> 4;       // lane half: K pair {0,1} vs {2,3}
  const int ln   = lane & 15;

  // Banded Gaussian operand chunks. For the f32 A-layout (lane = row,
  // VGPR0=K{0|2}, VGPR1=K{1|3}) this encodes A[i][k]=g[k-i]; the identical
  // per-lane values also encode G[k][n]=g[k-n] in the symmetric B-layout.
  v2f band[7];
#pragma unroll
  for (int c = 0; c < 7; ++c) {
    int kk = 4 * c + 2 * hi;
    int d0 = kk - ln, d1 = d0 + 1;
    band[c].x = (d0 >= 0 && d0 <= 10) ? GW[d0] : 0.0f;
    band[c].y = (d1 >= 0 && d1 <= 10) ? GW[d1] : 0.0f;
  }

  v8f out[5];
#pragma unroll
  for (int m = 0; m < 5; ++m) {
    // Stage V: V(16x26) = A_band x M(26x26), split into two 16-col halves,
    // K padded 26 -> 28 (7 chunks of 4). Loads are unconditional (clamped
    // row index); K/N padding applied as select-to-zero on the data.
    v8f a0 = {0.f, 0.f, 0.f, 0.f, 0.f, 0.f, 0.f, 0.f};
    v8f a1 = a0;
#pragma unroll
    for (int c = 0; c < 7; ++c) {
      int kk = 4 * c + 2 * hi;
      int k0 = min(kk, 25), k1 = min(kk + 1, 25);
      bool v0 = kk < 26, v1 = (kk + 1) < 26;
      // half 0: tile cols 0..15
      float p1x = Y1s[r0 + k0][c0 + ln];
      float p2x = Y2s[r0 + k0][c0 + ln];
      float p1y = Y1s[r0 + k1][c0 + ln];
      float p2y = Y2s[r0 + k1][c0 + ln];
      p1x = v0 ? p1x : 0.f;  p2x = v0 ? p2x : 0.f;
      p1y = v1 ? p1y : 0.f;  p2y = v1 ? p2y : 0.f;
      v2f b0 = {map_val(m, p1x, p2x), map_val(m, p1y, p2y)};
      a0 = wmma4_f32(band[c], b0, a0);
      // half 1: tile cols 16..25 (26..31 zero-padded)
      int cc = c0 + 16 + min(ln, 9);
      bool vn = ln < 10;
      float q1x = Y1s[r0 + k0][cc];
      float q2x = Y2s[r0 + k0][cc];
      float q1y = Y1s[r0 + k1][cc];
      float q2y = Y2s[r0 + k1][cc];
      q1x = (v0 && vn) ? q1x : 0.f;  q2x = (v0 && vn) ? q2x : 0.f;
      q1y = (v1 && vn) ? q1y : 0.f;  q2y = (v1 && vn) ? q2y : 0.f;
      v2f b1 = {map_val(m, q1x, q2x), map_val(m, q1y, q2y)};
      a1 = wmma4_f32(band[c], b1, a1);
    }
    // Re-layout D-format -> A-format through per-wave LDS (same-wave DS
    // ordering; compiler inserts s_wait_dscnt before dependent reads).
#pragma unroll
    for (int v = 0; v < 8; ++v) {
      Vb[w][v + 8 * hi][ln]      = a0[v];
      Vb[w][v + 8 * hi][16 + ln] = a1[v];
    }
    // Stage H: Out(16x16) = V(16x28) x G_band(28x16)
    v8f o = {0.f, 0.f, 0.f, 0.f, 0.f, 0.f, 0.f, 0.f};
#pragma unroll
    for (int c = 0; c < 7; ++c) {
      int kk = 4 * c + 2 * hi;
      v2f av = {Vb[w][ln][kk], Vb[w][ln][kk + 1]};
      o = wmma4_f32(av, band[c], o);
    }
    out[m] = o;
  }

  // ---- Phase 3: SSIM map in C-layout, masked mean contribution ----
  float lsum = 0.0f;
#pragma unroll
  for (int v = 0; v < 8; ++v) {
    int orow = Ry + r0 + v + 8 * hi;
    int ocol = Rx + c0 + ln;
    float mu1 = out[0][v], mu2 = out[1][v];
    float e11 = out[2][v], e22 = out[3][v], e12 = out[4][v];
    float mu1s = mu1 * mu1, mu2s = mu2 * mu2, mu12 = mu1 * mu2;
    float s11 = e11 - mu1s, s22 = e22 - mu2s, s12 = e12 - mu12;
    float num = (2.0f * mu12 + SSIM_C1) * (2.0f * s12 + SSIM_C2);
    float den = (mu1s + mu2s + SSIM_C1) * (s11 + s22 + SSIM_C2);
    float val = num / den;
    lsum += (orow < 502 && ocol < 502) ? val : 0.f;
  }
  // wave32 butterfly reduction
  for (int off = 16; off > 0; off >>= 1) lsum += __shfl_xor(lsum, off, 32);
  if (lane == 0) atomicAdd(acc, (double)lsum);
}

__global__ void ssim_finalize(const double* __restrict__ acc,
                              float* __restrict__ out) {
  if (threadIdx.x == 0 && blockIdx.x == 0) {
    out[0] = (float)(acc[0] * (1.0 / 8064128.0));  // 32 * 502 * 502
  }
}

extern "C" void kernel_launch(void* const* d_in, const int* in_sizes, int n_in,
                              void* d_out, int out_size, void* d_ws,
                              size_t ws_size, hipStream_t stream) {
  (void)in_sizes; (void)n_in; (void)out_size; (void)ws_size;
  const float* img1 = (const float*)d_in[0];
  const float* img2 = (const float*)d_in[1];
  float* out = (float*)d_out;
  double* acc = (double*)d_ws;

  ssim_zero_acc<<<1, 1, 0, stream>>>(acc);
  // 502 valid rows/cols: ceil(502/64)=8 region-rows, ceil(502/32)=16 cols
  dim3 grid(16, 8, 32);
  ssim_main_kernel<<<grid, 256, 0, stream>>>(img1, img2, acc);
  ssim_finalize<<<1, 1, 0, stream>>>(acc, out);
}